// BatchedTrilLinear_55551107006844
// MI455X (gfx1250) — compile-verified
//
#include <hip/hip_runtime.h>
#include <math.h>

// --- Problem constants (from reference) ---
#define NROWS 4096          // samples
#define NBLK  512           // batches / blocks
#define DD    64            // block dim (in == out)
#define ROWSTRIDE (NBLK * DD)   // 32768 floats per x/y row
#define LDS_STRIDE 68           // 64 + 4 pad -> conflict-free b64 B-frag reads
#define WG_THREADS 128          // 4 wave32
#define ROWS_PER_WAVE 16
#define ROWS_PER_WG (4 * ROWS_PER_WAVE)   // 64

typedef __attribute__((ext_vector_type(2))) float v2f;
typedef __attribute__((ext_vector_type(8))) float v8f;

__global__ __launch_bounds__(WG_THREADS)
void BatchedTrilLinear_wmma_f32(const float* __restrict__ x,
                                const float* __restrict__ w,
                                float* __restrict__ y)
{
    __shared__ float ldsL[DD * LDS_STRIDE];   // L_b row-major [o][i], padded

    const int b     = blockIdx.x;     // which 64x64 block
    const int chunk = blockIdx.y;     // which 64-row slab of x
    const int tid   = threadIdx.x;
    const int lane  = tid & 31;
    const int wave  = tid >> 5;

    // ---- Stage L_b = tril(W_b,-1) + diag(exp(diag W_b)) into LDS ----
    const float* wb = w + (size_t)b * DD * DD;
    #pragma unroll
    for (int base = 0; base < DD * DD; base += WG_THREADS) {
        int idx = base + tid;
        int o = idx >> 6;        // output row
        int i = idx & 63;        // input col
        float v  = wb[idx];
        float lv = (i < o) ? v : ((i == o) ? __expf(v) : 0.0f);
        ldsL[o * LDS_STRIDE + i] = lv;
    }
    __syncthreads();

    // ---- Per-wave 16x64 output tile via V_WMMA_F32_16X16X4_F32 ----
    // A (16x4 f32): lanes 0-15 hold M=lane, K={k,k+1}; lanes 16-31 hold K={k+2,k+3}
    const int m16   = lane & 15;
    const int khalf = (lane >> 4) << 1;            // 0 or 2
    const int row0  = chunk * ROWS_PER_WG + wave * ROWS_PER_WAVE;

    const float* xrow = x + (size_t)(row0 + m16) * ROWSTRIDE + b * DD;
    // B (4x16): B[k][n] = L[n][k]  (Y = X * L^T); lane holds n = n0 + m16,
    // v0/v1 = L[n][kbase+khalf], L[n][kbase+khalf+1]  -> contiguous b64 in LDS row n
    const float* lrow = &ldsL[m16 * LDS_STRIDE + khalf];

    v8f c0 = {}, c1 = {}, c2 = {}, c3 = {};

    #pragma unroll 4
    for (int k = 0; k < DD; k += 4) {
        v2f a  = *(const v2f*)(xrow + k + khalf);
        const float* lb = lrow + k;
        v2f b0 = *(const v2f*)(lb +  0 * LDS_STRIDE);
        v2f b1 = *(const v2f*)(lb + 16 * LDS_STRIDE);
        v2f b2 = *(const v2f*)(lb + 32 * LDS_STRIDE);
        v2f b3 = *(const v2f*)(lb + 48 * LDS_STRIDE);
        // 8 args: (neg_a, A, neg_b, B, c_mod, C, reuse_a, reuse_b)
        c0 = __builtin_amdgcn_wmma_f32_16x16x4_f32(false, a, false, b0, (short)0, c0, false, false);
        c1 = __builtin_amdgcn_wmma_f32_16x16x4_f32(false, a, false, b1, (short)0, c1, false, false);
        c2 = __builtin_amdgcn_wmma_f32_16x16x4_f32(false, a, false, b2, (short)0, c2, false, false);
        c3 = __builtin_amdgcn_wmma_f32_16x16x4_f32(false, a, false, b3, (short)0, c3, false, false);
    }

    // ---- Store: C/D VGPR j -> row (j + 8*(lane>=16)), col = lane&15 ----
    float* ybase = y + (size_t)row0 * ROWSTRIDE + b * DD + m16;
    const int rsub = (lane >> 4) << 3;   // 0 or 8
    #pragma unroll
    for (int j = 0; j < 8; ++j) {
        float* yr = ybase + (size_t)(rsub + j) * ROWSTRIDE;
        yr[ 0] = c0[j];
        yr[16] = c1[j];
        yr[32] = c2[j];
        yr[48] = c3[j];
    }
}

extern "C" void kernel_launch(void* const* d_in, const int* in_sizes, int n_in,
                              void* d_out, int out_size, void* d_ws, size_t ws_size,
                              hipStream_t stream) {
    (void)in_sizes; (void)n_in; (void)out_size; (void)d_ws; (void)ws_size;
    const float* x = (const float*)d_in[0];     // [4096, 32768] f32
    const float* w = (const float*)d_in[1];     // [512, 64, 64] f32
    float* y = (float*)d_out;                   // [4096, 32768] f32

    dim3 grid(NBLK, NROWS / ROWS_PER_WG);       // (512, 64)
    BatchedTrilLinear_wmma_f32<<<grid, WG_THREADS, 0, stream>>>(x, w, y);
}